// ElectricField_55284819034161
// MI455X (gfx1250) — compile-verified
//
#include <hip/hip_runtime.h>

// ElectricField scatter for MI455X (gfx1250, wave32).
// HBM stream: 24 B/edge (NT b128 loads, 4 edges/thread); gathers + f32 atomic
// scatter stay L2-resident. Raw v_log/v_exp/v_sqrt/v_rcp keep VALU count low
// so the kernel stays at the HBM roofline instead of going issue-bound.

typedef float v4f __attribute__((ext_vector_type(4)));
typedef int   v4i __attribute__((ext_vector_type(4)));

// -0.7 * log2(e): expf(-0.7*x) == exp2f(EXP2_SCALE * x)
#define EXP2_SCALE (-1.0098865309f)

__global__ __launch_bounds__(256) void ef_zero_kernel(float* __restrict__ out, int n) {
    int i = blockIdx.x * blockDim.x + threadIdx.x;
    if (i < n) out[i] = 0.0f;
}

__device__ __forceinline__ void ef_one_edge(
    int s, int d, float r, float vx, float vy, float vz,
    const float* __restrict__ charges, const float* __restrict__ polar,
    float* __restrict__ out)
{
    float q  = charges[d];                       // L2-resident gathers
    float ps = polar[s];
    float pd = polar[d];

    float alpha = ps * pd;                       // in [0.25, 6.25] -> safe for raw ops
    float l     = __builtin_amdgcn_logf(alpha);  // log2(alpha)
    // uij^1.5 = (r * alpha^(-1/6))^1.5 = r^1.5 * alpha^(-1/4)
    float B     = __builtin_amdgcn_exp2f(-0.25f * l);
    float sr    = __builtin_amdgcn_sqrtf(r);     // r in [0.5, 4.5]
    float u15   = (r * sr) * B;
    float E     = __builtin_amdgcn_exp2f(EXP2_SCALE * u15);  // exp(-0.7*u15)
    float r3    = (r * r) * r;
    float qr    = q * __builtin_amdgcn_rcpf(r3);
    // coeff = -q*(1-E)/r^3 = qr*E - qr
    float coeff = __builtin_fmaf(qr, E, -qr);

    float* o = out + 3ll * s;
    __hip_atomic_fetch_add(o + 0, coeff * vx, __ATOMIC_RELAXED, __HIP_MEMORY_SCOPE_AGENT);
    __hip_atomic_fetch_add(o + 1, coeff * vy, __ATOMIC_RELAXED, __HIP_MEMORY_SCOPE_AGENT);
    __hip_atomic_fetch_add(o + 2, coeff * vz, __ATOMIC_RELAXED, __HIP_MEMORY_SCOPE_AGENT);
}

__global__ __launch_bounds__(256) void ef_edge_kernel(
    const int*   __restrict__ edge_src,
    const int*   __restrict__ edge_dst,
    const float* __restrict__ dist,
    const float* __restrict__ vec,      // [E,3] row-major
    const float* __restrict__ charges,  // [N]
    const float* __restrict__ polar,    // [N]
    float*       __restrict__ out,      // [N,3] zeroed
    int nE)
{
    long t    = (long)blockIdx.x * blockDim.x + threadIdx.x;
    long base = 4l * t;
    if (base + 3 < nE) {
        // 128-bit non-temporal streamed loads: 6 vmem ops cover 4 edges.
        v4i s4 = __builtin_nontemporal_load((const v4i*)(edge_src + base));
        v4i d4 = __builtin_nontemporal_load((const v4i*)(edge_dst + base));
        v4f r4 = __builtin_nontemporal_load((const v4f*)(dist + base));
        const v4f* vp = (const v4f*)(vec + 3 * base);   // 48*t bytes -> 16B aligned
        v4f va = __builtin_nontemporal_load(vp + 0);
        v4f vb = __builtin_nontemporal_load(vp + 1);
        v4f vc = __builtin_nontemporal_load(vp + 2);

        ef_one_edge(s4.x, d4.x, r4.x, va.x, va.y, va.z, charges, polar, out);
        ef_one_edge(s4.y, d4.y, r4.y, va.w, vb.x, vb.y, charges, polar, out);
        ef_one_edge(s4.z, d4.z, r4.z, vb.z, vb.w, vc.x, charges, polar, out);
        ef_one_edge(s4.w, d4.w, r4.w, vc.y, vc.z, vc.w, charges, polar, out);
    } else {
        // Scalar tail (only if nE % 4 != 0; nE = 6.4M is exact).
        for (long e = base; e < nE; ++e) {
            int   s  = edge_src[e];
            int   d  = edge_dst[e];
            float r  = dist[e];
            const float* vv = vec + 3 * e;
            ef_one_edge(s, d, r, vv[0], vv[1], vv[2], charges, polar, out);
        }
    }
}

extern "C" void kernel_launch(void* const* d_in, const int* in_sizes, int n_in,
                              void* d_out, int out_size, void* d_ws, size_t ws_size,
                              hipStream_t stream) {
    // Input order: species, edge_src, edge_dst, distances, vec, charges, polarisability
    const int*   edge_src = (const int*)  d_in[1];
    const int*   edge_dst = (const int*)  d_in[2];
    const float* dist     = (const float*)d_in[3];
    const float* vec      = (const float*)d_in[4];
    const float* charges  = (const float*)d_in[5];
    const float* polar    = (const float*)d_in[6];
    float*       out      = (float*)d_out;

    int nE = in_sizes[1];

    {   // zero the accumulator (harness poisons d_out)
        int threads = 256;
        int blocks  = (out_size + threads - 1) / threads;
        ef_zero_kernel<<<blocks, threads, 0, stream>>>(out, out_size);
    }
    {   // 4 edges per lane; 256 threads = 8 wave32 per block
        int threads = 256;
        long edgesPerBlock = 4l * threads;
        int blocks = (int)((nE + edgesPerBlock - 1) / edgesPerBlock);
        ef_edge_kernel<<<blocks, threads, 0, stream>>>(
            edge_src, edge_dst, dist, vec, charges, polar, out, nE);
    }
}